// SSM_60902636257786
// MI455X (gfx1250) — compile-verified
//
#include <hip/hip_runtime.h>
#include <math.h>

#define BATCH   16
#define C_IN    64
#define LSEQ    4096
#define D_MODEL 256
#define NSTATE  64
#define OUTC    10

typedef __attribute__((ext_vector_type(2))) float v2f;
typedef __attribute__((ext_vector_type(8))) float v8f;

// Workspace layout (floats):
//   [0, 4096)        w[k]
//   [4096]           s = mean(w)
//   [4224, 5248)     v  (16 x 64)
//   [8192, 12288)    y  (16 x 256)
#define WS_W 0
#define WS_S 4096
#define WS_V 4224
#define WS_Y 8192

// ---------------------------------------------------------------------------
// Kernel 1: closed-form time weights.
//   w[k] = sum_i Cv[i]*Bv[i] * (1 - a_i^(L-k)) / (1 - a_i),  a_i = A[i,i]
// Fully parallel over k (no serial 4096-step scan).
// ---------------------------------------------------------------------------
__global__ void ssm_weights_kernel(const float* __restrict__ A,
                                   const float* __restrict__ Bv,
                                   const float* __restrict__ Cv,
                                   float* __restrict__ ws) {
  int k = blockIdx.x * blockDim.x + threadIdx.x;   // 0..4095
  if (k >= LSEQ) return;
  float p = (float)(LSEQ - k);
  float acc = 0.f;
  #pragma unroll 4
  for (int i = 0; i < NSTATE; ++i) {
    float a  = A[i * NSTATE + i];        // diagonal entry
    float cb = Cv[i] * Bv[i];
    float om = 1.0f - a;
    float g  = (om > 1e-6f) ? (1.0f - __powf(a, p)) / om : p;  // geometric sum
    acc = fmaf(cb, g, acc);
  }
  ws[WS_W + k] = acc;
}

// ---------------------------------------------------------------------------
// Kernel 2: s = (1/L) * sum_k w[k]   (single block reduction)
// ---------------------------------------------------------------------------
__global__ void ssm_s_kernel(float* __restrict__ ws) {
  __shared__ float sh[256];
  int t = threadIdx.x;
  float acc = 0.f;
  for (int k = t; k < LSEQ; k += 256) acc += ws[WS_W + k];
  sh[t] = acc;
  __syncthreads();
  for (int off = 128; off > 0; off >>= 1) {
    if (t < off) sh[t] += sh[t + off];
    __syncthreads();
  }
  if (t == 0) ws[WS_S] = sh[0] * (1.0f / (float)LSEQ);
}

// ---------------------------------------------------------------------------
// Kernel 3: v[b,c] = (1/L) * sum_k w[k] * u[b,c,k]
// One block per (b,c) row; float4 (b128) streaming loads. This is the only
// kernel that touches the 16.8 MB of u -> pure bandwidth.
// ---------------------------------------------------------------------------
__global__ void ssm_reduce_u_kernel(const float* __restrict__ u,
                                    const float* __restrict__ ws_w,
                                    float* __restrict__ v) {
  int row = blockIdx.x;                              // b*C_IN + c, 0..1023
  int t   = threadIdx.x;                             // 0..255
  const float4* u4 = (const float4*)(u + (size_t)row * LSEQ);
  const float4* w4 = (const float4*)ws_w;
  float acc = 0.f;
  #pragma unroll
  for (int it = 0; it < (LSEQ / 4) / 256; ++it) {
    int idx = t + it * 256;
    float4 uu = u4[idx];
    float4 wv = w4[idx];
    acc = fmaf(uu.x, wv.x, acc);
    acc = fmaf(uu.y, wv.y, acc);
    acc = fmaf(uu.z, wv.z, acc);
    acc = fmaf(uu.w, wv.w, acc);
  }
  __shared__ float sh[256];
  sh[t] = acc;
  __syncthreads();
  for (int off = 128; off > 0; off >>= 1) {
    if (t < off) sh[t] += sh[t + off];
    __syncthreads();
  }
  if (t == 0) v[row] = sh[0] * (1.0f / (float)LSEQ);
}

// ---------------------------------------------------------------------------
// Kernel 4: y = v @ W_up^T + s*b_up    (16x256 = [16x64]*[64x256])
// One wave per 16-wide d-tile, V_WMMA_F32_16X16X4_F32, K=64 -> 16 WMMAs.
// A-frag: lane m=lane%16, VGPR0/1 hold K = half..half+1 (half = 0 or 2).
// B-frag: lane n=lane%16, same K striping. C/D: VGPR r -> row r (+8 upper).
// ---------------------------------------------------------------------------
__global__ void ssm_gemm_y_wmma(const float* __restrict__ v,
                                const float* __restrict__ W_up,
                                const float* __restrict__ b_up,
                                const float* __restrict__ ws,
                                float* __restrict__ y) {
  int lane = threadIdx.x;                 // 0..31 (one wave32)
  int mn   = lane & 15;
  int half = (lane < 16) ? 0 : 2;
  int n0   = blockIdx.x * 16;             // d tile base
  float s  = ws[WS_S];

  v8f acc = {};
  #pragma unroll
  for (int k0 = 0; k0 < C_IN; k0 += 4) {
    v2f a, b;
    a.x = v[mn * C_IN + k0 + half];                 // A = v (M=16 rows=batch)
    a.y = v[mn * C_IN + k0 + half + 1];
    b.x = W_up[(n0 + mn) * C_IN + k0 + half];       // B[k][n] = W_up[n0+n, k]
    b.y = W_up[(n0 + mn) * C_IN + k0 + half + 1];
    acc = __builtin_amdgcn_wmma_f32_16x16x4_f32(false, a, false, b,
                                                (short)0, acc, false, false);
  }
  float bias = s * b_up[n0 + mn];
  #pragma unroll
  for (int r = 0; r < 8; ++r) {
    int m = r + ((lane < 16) ? 0 : 8);
    y[m * D_MODEL + n0 + mn] = acc[r] + bias;
  }
}

// ---------------------------------------------------------------------------
// Kernel 5: out = y @ W_down^T + b_down   (16x10, padded to N=16)
// Single wave, K=256 -> 64 WMMAs. Columns >= 10 are fed zeros (clamped
// address + mask, no OOB reads) and never stored.
// ---------------------------------------------------------------------------
__global__ void ssm_gemm_out_wmma(const float* __restrict__ y,
                                  const float* __restrict__ W_down,
                                  const float* __restrict__ b_down,
                                  float* __restrict__ out) {
  int lane = threadIdx.x;
  int mn   = lane & 15;
  int half = (lane < 16) ? 0 : 2;
  float msk = (mn < OUTC) ? 1.0f : 0.0f;
  int   nc  = (mn < OUTC) ? mn : 0;       // clamp to stay in-bounds

  v8f acc = {};
  #pragma unroll
  for (int k0 = 0; k0 < D_MODEL; k0 += 4) {
    v2f a, b;
    a.x = y[mn * D_MODEL + k0 + half];
    a.y = y[mn * D_MODEL + k0 + half + 1];
    b.x = W_down[nc * D_MODEL + k0 + half] * msk;   // B[k][n] = W_down[n, k]
    b.y = W_down[nc * D_MODEL + k0 + half + 1] * msk;
    acc = __builtin_amdgcn_wmma_f32_16x16x4_f32(false, a, false, b,
                                                (short)0, acc, false, false);
  }
  if (mn < OUTC) {
    float bias = b_down[mn];
    #pragma unroll
    for (int r = 0; r < 8; ++r) {
      int m = r + ((lane < 16) ? 0 : 8);
      out[m * OUTC + mn] = acc[r] + bias;
    }
  }
}

// ---------------------------------------------------------------------------
extern "C" void kernel_launch(void* const* d_in, const int* in_sizes, int n_in,
                              void* d_out, int out_size, void* d_ws, size_t ws_size,
                              hipStream_t stream) {
  const float* u      = (const float*)d_in[0];   // (16,64,4096)
  const float* W_up   = (const float*)d_in[1];   // (256,64)
  const float* b_up   = (const float*)d_in[2];   // (256,)
  const float* A      = (const float*)d_in[3];   // (64,64) diagonal
  const float* Bv     = (const float*)d_in[4];   // (64,)
  const float* Cv     = (const float*)d_in[5];   // (64,)
  const float* W_down = (const float*)d_in[6];   // (10,256)
  const float* b_down = (const float*)d_in[7];   // (10,)
  float* out = (float*)d_out;                    // (16,10)
  float* ws  = (float*)d_ws;

  // 1) time weights w[k] (closed form, parallel over k)
  ssm_weights_kernel<<<LSEQ / 256, 256, 0, stream>>>(A, Bv, Cv, ws);
  // 2) s = mean(w)
  ssm_s_kernel<<<1, 256, 0, stream>>>(ws);
  // 3) v[b,c] = (1/L) sum_k w[k] u[b,c,k]   (the only big memory pass)
  ssm_reduce_u_kernel<<<BATCH * C_IN, 256, 0, stream>>>(u, ws + WS_W, ws + WS_V);
  // 4) y = v @ W_up^T + s*b_up  (WMMA f32 16x16x4, 16 waves)
  ssm_gemm_y_wmma<<<D_MODEL / 16, 32, 0, stream>>>(ws + WS_V, W_up, b_up, ws, ws + WS_Y);
  // 5) out = y @ W_down^T + b_down  (WMMA, single wave)
  ssm_gemm_out_wmma<<<1, 32, 0, stream>>>(ws + WS_Y, W_down, b_down, out);
}